// HATD_20736102105419
// MI455X (gfx1250) — compile-verified
//
#include <hip/hip_runtime.h>
#include <stdint.h>

// Problem constants (from reference setup_inputs)
#define BB 4
#define HH 8
#define NN 2048
#define CC 512
#define NSQ ((size_t)NN * (size_t)NN)
#define TAU 1.5f
#define EPS_E 1e-6f
#define MIN_TOKENS 32

// ---------------------------------------------------------------------------
// CDNA5 feature detection
// ---------------------------------------------------------------------------
#if defined(__has_builtin)
#if __has_builtin(__builtin_amdgcn_global_load_async_to_lds_b128) && \
    __has_builtin(__builtin_amdgcn_s_wait_asynccnt)
#define HAVE_ASYNC_LDS 1
#endif
#if __has_builtin(__builtin_amdgcn_wmma_f32_16x16x4_f32)
#define HAVE_WMMA_F32 1
#endif
#endif

typedef float v2f __attribute__((ext_vector_type(2)));
typedef float v8f __attribute__((ext_vector_type(8)));
typedef int   v4i __attribute__((ext_vector_type(4)));

#if defined(HAVE_ASYNC_LDS)
typedef __attribute__((address_space(1))) v4i gv4i;  // global
typedef __attribute__((address_space(3))) v4i lv4i;  // LDS

__device__ __forceinline__ void cp_b128_to_lds(void* lds, const void* g) {
  __builtin_amdgcn_global_load_async_to_lds_b128(
      (gv4i*)(uintptr_t)g, (lv4i*)(uint32_t)(uintptr_t)lds, 0, 0);
}
#define WAIT_ASYNC(n) __builtin_amdgcn_s_wait_asynccnt(n)
#else
__device__ __forceinline__ void cp_b128_to_lds(void* lds, const void* g) {
  *(float4*)lds = *(const float4*)g;  // synchronous fallback
}
#define WAIT_ASYNC(n) ((void)0)
#endif

// ---------------------------------------------------------------------------
// Kernel 1: fused head-mean + entropy.  One block per (b, i) row.
// Streams 8 head-rows (8 KB each) through a double-buffered LDS stage fed by
// async global->LDS copies; per-thread accumulation; WMMA block reduction.
// ---------------------------------------------------------------------------
__global__ __launch_bounds__(256) void entropy_kernel(
    const float* __restrict__ attn, float* __restrict__ entropy) {
  __shared__ __align__(16) float buf[2][NN];   // 2 x 8KB staging
  __shared__ __align__(16) float P[16][16];    // partials for WMMA reduction

  const int bi = blockIdx.x;            // 0 .. B*N-1
  const int b  = bi >> 11;              // / 2048
  const int i  = bi & (NN - 1);
  const int t  = threadIdx.x;

  const float* base = attn + (size_t)b * HH * NSQ + (size_t)i * NN;

  // stage s == head s; each thread copies two b128 chunks (cols 4t, 1024+4t)
  auto issue = [&](int s) {
    const float* hp = base + (size_t)s * NSQ;
    float* dst = &buf[s & 1][0];
    cp_b128_to_lds(&dst[4 * t],        hp + 4 * t);
    cp_b128_to_lds(&dst[1024 + 4 * t], hp + 1024 + 4 * t);
  };

  float a0 = 0.f, a1 = 0.f, a2 = 0.f, a3 = 0.f;   // cols 4t..4t+3
  float b0 = 0.f, b1 = 0.f, b2 = 0.f, b3 = 0.f;   // cols 1024+4t..+3

  issue(0);
  for (int h = 0; h < HH; ++h) {
    if (h + 1 < HH) {
      issue(h + 1);
      WAIT_ASYNC(2);   // stage h complete (its 2 instrs retired, in order)
    } else {
      WAIT_ASYNC(0);
    }
    __syncthreads();      // all waves' async writes now visible
    const float* s = buf[h & 1];
    float4 va = *(const float4*)&s[4 * t];
    float4 vb = *(const float4*)&s[1024 + 4 * t];
    a0 += va.x; a1 += va.y; a2 += va.z; a3 += va.w;
    b0 += vb.x; b1 += vb.y; b2 += vb.z; b3 += vb.w;
    __syncthreads();      // stage buffer may be reused next-next iteration
  }

  // mean over heads, then -sum p*log(p+eps) over this thread's 8 columns
  float part = 0.f;
  {
    float m;
    m = a0 * 0.125f; part -= m * __logf(m + EPS_E);
    m = a1 * 0.125f; part -= m * __logf(m + EPS_E);
    m = a2 * 0.125f; part -= m * __logf(m + EPS_E);
    m = a3 * 0.125f; part -= m * __logf(m + EPS_E);
    m = b0 * 0.125f; part -= m * __logf(m + EPS_E);
    m = b1 * 0.125f; part -= m * __logf(m + EPS_E);
    m = b2 * 0.125f; part -= m * __logf(m + EPS_E);
    m = b3 * 0.125f; part -= m * __logf(m + EPS_E);
  }

  // block reduction of 256 partials
  P[t >> 4][t & 15] = part;
  __syncthreads();

  if (t < 32) {  // wave 0 only; EXEC all ones within the wave
#if defined(HAVE_WMMA_F32)
    // Sum(P) via 4 chained V_WMMA_F32_16X16X4_F32 with B = ones(4x16).
    // A chunk layout (32-bit A 16x4): lanes 0-15 -> K=0,1 ; lanes 16-31 -> K=2,3
    const int r  = t & 15;
    const int hi = (t >= 16) ? 2 : 0;
    v2f ones; ones.x = 1.0f; ones.y = 1.0f;
    v8f acc = {};
    for (int c0 = 0; c0 < 16; c0 += 4) {
      v2f a;
      a.x = P[r][c0 + hi];
      a.y = P[r][c0 + hi + 1];
      acc = __builtin_amdgcn_wmma_f32_16x16x4_f32(
          false, a, false, ones, (short)0, acc, false, false);
    }
    // D[m][n] = rowsum_m (identical for every n). Lanes<16 hold m=0..7,
    // lanes>=16 hold m=8..15 across the 8 acc VGPRs.
    float s = acc[0] + acc[1] + acc[2] + acc[3] +
              acc[4] + acc[5] + acc[6] + acc[7];
    s += __shfl_xor(s, 16, 32);
#else
    const int r = t >> 1, cb = (t & 1) * 8;
    float s = 0.f;
    for (int k = 0; k < 8; ++k) s += P[r][cb + k];
    s += __shfl_xor(s, 16, 32);
    s += __shfl_xor(s, 8, 32);
    s += __shfl_xor(s, 4, 32);
    s += __shfl_xor(s, 2, 32);
    s += __shfl_xor(s, 1, 32);
#endif
    if (t == 0) entropy[bi] = s;
  }
}

// ---------------------------------------------------------------------------
// Kernel 2: per-batch selection. One block per batch.
// Produces idx[b][n] = source token (or -1 => zero row).
// ---------------------------------------------------------------------------
__global__ __launch_bounds__(256) void select_kernel(
    const float* __restrict__ entropy, int* __restrict__ idxOut) {
  __shared__ float ent[NN];      // 8 KB
  __shared__ float work[NN];     // 8 KB (destructible copy for top-k)
  __shared__ int   order[NN];    // 8 KB (stable compaction)
  __shared__ int   tsum[257];
  __shared__ float bestV[256];
  __shared__ int   bestI[256];
  __shared__ int   topk[MIN_TOKENS];

  const int b = blockIdx.x;
  const int t = threadIdx.x;
  const float* e = entropy + (size_t)b * NN;

  for (int n = t; n < NN; n += 256) {
    float v = e[n];
    ent[n] = v;
    work[n] = v;
  }
  __syncthreads();

  // ---- count + exclusive scan of mask (contiguous 8-element chunks) ----
  const int base = t * 8;
  int localCnt = 0;
  for (int k = 0; k < 8; ++k) localCnt += (ent[base + k] > TAU) ? 1 : 0;
  tsum[t] = localCnt;
  __syncthreads();
  if (t == 0) {
    int acc = 0;
    for (int j = 0; j < 256; ++j) { int v = tsum[j]; tsum[j] = acc; acc += v; }
    tsum[256] = acc;
  }
  __syncthreads();
  const int count = tsum[256];

  // stable compaction: kept tokens first (original order), dropped after.
  // exclusive-prefix of !mask at n == n - exclusive-prefix of mask at n
  {
    int kEx = tsum[t];
    for (int k = 0; k < 8; ++k) {
      const int n = base + k;
      const bool m = ent[n] > TAU;
      const int pos = m ? kEx : (count + (n - kEx));
      order[pos] = n;
      kEx += m ? 1 : 0;
    }
  }
  __syncthreads();

  // ---- top-32 by entropy, descending; ties -> lower index first ----
  for (int it = 0; it < MIN_TOKENS; ++it) {
    float bv = -__builtin_inff();
    int bidx = NN;
    for (int k = 0; k < 8; ++k) {
      const int n = base + k;
      const float v = work[n];
      if (v > bv || (v == bv && n < bidx)) { bv = v; bidx = n; }
    }
    bestV[t] = bv;
    bestI[t] = bidx;
    __syncthreads();
    if (t == 0) {
      float BV = -__builtin_inff();
      int BI = NN;
      for (int j = 0; j < 256; ++j) {
        if (bestV[j] > BV || (bestV[j] == BV && bestI[j] < BI)) {
          BV = bestV[j];
          BI = bestI[j];
        }
      }
      topk[it] = BI;
      work[BI] = -__builtin_inff();
    }
    __syncthreads();
  }

  const bool use_topk = count < MIN_TOKENS;
  const int klen = use_topk ? MIN_TOKENS : count;

  for (int n = t; n < NN; n += 256) {
    int v = -1;
    if (n < klen) v = use_topk ? topk[n] : order[n];
    idxOut[(size_t)b * NN + n] = v;
  }
}

// ---------------------------------------------------------------------------
// Kernel 3: gather + zero-pad. One block per output row, float4 per thread.
// ---------------------------------------------------------------------------
__global__ __launch_bounds__(128) void gather_kernel(
    const float* __restrict__ tokens, const int* __restrict__ idx,
    float* __restrict__ out) {
  const int bn = blockIdx.x;          // 0 .. B*N-1
  const int b = bn >> 11;
  const int t = threadIdx.x;          // 128 threads * float4 = 512 floats
  const int src = idx[bn];
  float4* o = (float4*)(out + (size_t)bn * CC);
  if (src < 0) {
    float4 z; z.x = 0.f; z.y = 0.f; z.z = 0.f; z.w = 0.f;
    o[t] = z;
  } else {
    const float4* s =
        (const float4*)(tokens + ((size_t)b * NN + (size_t)src) * CC);
    o[t] = s[t];
  }
}

// ---------------------------------------------------------------------------
// Host entry
// ---------------------------------------------------------------------------
extern "C" void kernel_launch(void* const* d_in, const int* in_sizes, int n_in,
                              void* d_out, int out_size, void* d_ws,
                              size_t ws_size, hipStream_t stream) {
  const float* tokens = (const float*)d_in[0];  // (B, N, C)
  const float* attn   = (const float*)d_in[1];  // (B, H, N, N)
  float* out = (float*)d_out;                   // (B, N, C)

  char* ws = (char*)d_ws;
  float* ent_ws = (float*)ws;                            // B*N floats (32 KB)
  int*   idx_ws = (int*)(ws + sizeof(float) * BB * NN);  // B*N ints (32 KB)

  entropy_kernel<<<dim3(BB * NN), dim3(256), 0, stream>>>(attn, ent_ws);
  select_kernel<<<dim3(BB), dim3(256), 0, stream>>>(ent_ws, idx_ws);
  gather_kernel<<<dim3(BB * NN), dim3(128), 0, stream>>>(tokens, idx_ws, out);
}